// MOLAttn_60962765800017
// MI455X (gfx1250) — compile-verified
//
#include <hip/hip_runtime.h>
#include <hip/hip_bf16.h>

// ---------------------------------------------------------------------------
// MOLAttn on gfx1250 (MI455X).
// Dims fixed by the reference: B=1024, H=512, K=5, RNN_IN=512, P_HID=256, S=8192.
// ---------------------------------------------------------------------------

#define BDIM   1024
#define HDIM   512
#define KMIX   5
#define PHID   256
#define SLEN   8192

typedef __attribute__((ext_vector_type(2))) float v2f;
typedef __attribute__((ext_vector_type(8))) float v8f;

__device__ __forceinline__ float sigm(float x) {
    return 1.0f / (1.0f + __expf(-x));
}

// ---------------------------------------------------------------------------
// WMMA GEMM: C[M x N] = X[M x 512] * W[N x 512]^T + bias[N]  (optional tanh)
// Block = 256 threads = 8 waves.  Block tile: 64(M) x 64(N).
// Wave w: wm = w&3 -> 16-row M tile, wn = w>>2 -> two 16-col N tiles (32 cols).
// K staged 16 wide through LDS; LDS row stride 20 floats:
//   - float4 tile stores stay 16B aligned (20*4 = 80 bytes, kc in {0,4,8,12})
//   - fragment reads (addr = row*20 + kk) hit distinct banks across 16 lanes.
// Uses V_WMMA_F32_16X16X4_F32 (full f32, matches reference numerics).
// ---------------------------------------------------------------------------
__global__ __launch_bounds__(256)
void gemm_f32_wmma(const float* __restrict__ X,
                   const float* __restrict__ W,
                   const float* __restrict__ bias,
                   float* __restrict__ C,
                   int N, int act_tanh) {
    __shared__ float Xs[64 * 20];
    __shared__ float Ws[64 * 20];

    const int tid  = threadIdx.x;
    const int lane = tid & 31;
    const int w    = tid >> 5;          // wave id 0..7 (wave32)
    const int wm   = w & 3;             // M tile within block (0..3)
    const int wn   = w >> 2;            // N tile pair within block (0..1)
    const int half = lane >> 4;         // 0: lanes 0-15, 1: lanes 16-31
    const int lm   = lane & 15;

    const int n0 = blockIdx.x * 64;
    const int m0 = blockIdx.y * 64;

    // tile-load mapping: each thread moves one float4 per array per K step
    const int lrow = tid >> 2;          // 0..63
    const int lcol = (tid & 3) * 4;     // 0,4,8,12

    v8f c0 = {};
    v8f c1 = {};

    for (int k0 = 0; k0 < HDIM; k0 += 16) {
        __syncthreads();
        float4 xv = *(const float4*)(X + (m0 + lrow) * HDIM + k0 + lcol);
        float4 wv = *(const float4*)(W + (n0 + lrow) * HDIM + k0 + lcol);
        *(float4*)(&Xs[lrow * 20 + lcol]) = xv;
        *(float4*)(&Ws[lrow * 20 + lcol]) = wv;
        __syncthreads();

#pragma unroll
        for (int kk = 0; kk < 16; kk += 4) {
            // A fragment (16x4 f32): lanes 0-15 hold K=kk,kk+1; lanes 16-31 K=kk+2,kk+3
            v2f a;
            a.x = Xs[(wm * 16 + lm) * 20 + kk + half * 2];
            a.y = Xs[(wm * 16 + lm) * 20 + kk + half * 2 + 1];
            // B fragments (4x16): B[k][n] = W[n][k]
            v2f b0, b1;
            b0.x = Ws[(wn * 32 + lm) * 20 + kk + half * 2];
            b0.y = Ws[(wn * 32 + lm) * 20 + kk + half * 2 + 1];
            b1.x = Ws[(wn * 32 + 16 + lm) * 20 + kk + half * 2];
            b1.y = Ws[(wn * 32 + 16 + lm) * 20 + kk + half * 2 + 1];

            c0 = __builtin_amdgcn_wmma_f32_16x16x4_f32(
                     false, a, false, b0, (short)0, c0, false, false);
            c1 = __builtin_amdgcn_wmma_f32_16x16x4_f32(
                     false, a, false, b1, (short)0, c1, false, false);
        }
    }

    // Epilogue: C/D layout -> VGPR v: lanes 0-15 => M=v, lanes 16-31 => M=v+8
#pragma unroll
    for (int t = 0; t < 2; ++t) {
        const int col = n0 + wn * 32 + t * 16 + lm;
        const float bv = bias[col];
#pragma unroll
        for (int v = 0; v < 8; ++v) {
            const int row = m0 + wm * 16 + half * 8 + v;
            float val = (t == 0 ? c0[v] : c1[v]) + bv;
            if (act_tanh) val = tanhf(val);
            C[row * N + col] = val;
        }
    }
}

// ---------------------------------------------------------------------------
// GRU gate fuse: h = (1-z)*n from gi (= x@w_ih^T + b_ih) and b_hh.
// Writes h straight into the `state` output slice of d_out.
// ---------------------------------------------------------------------------
__global__ __launch_bounds__(256)
void gru_gate(const float* __restrict__ gi,
              const float* __restrict__ b_hh,
              float* __restrict__ h_out) {
    const int i = blockIdx.x * blockDim.x + threadIdx.x;   // < B*512
    const int b = i >> 9;
    const int j = i & 511;
    const float* g = gi + b * (3 * HDIM);
    const float r = sigm(g[j]           + b_hh[j]);
    const float z = sigm(g[HDIM + j]    + b_hh[HDIM + j]);
    const float n = tanhf(g[2 * HDIM + j] + r * b_hh[2 * HDIM + j]);
    h_out[i] = (1.0f - z) * n;
}

// ---------------------------------------------------------------------------
// params[b,k,q] = sum_p hid[b, k*256+p] * p_w2[k,q,p] + p_b2[k,q]
// One thread per (b,k,q): B*K*3 = 15360 threads, 256-long dot each.
// ---------------------------------------------------------------------------
__global__ __launch_bounds__(256)
void proj_params(const float* __restrict__ hid,
                 const float* __restrict__ p_w2,
                 const float* __restrict__ p_b2,
                 float* __restrict__ par) {
    const int idx = blockIdx.x * blockDim.x + threadIdx.x;  // < B*15
    const int q = idx % 3;
    const int k = (idx / 3) % KMIX;
    const int b = idx / 15;
    const float* hp = hid + b * (KMIX * PHID) + k * PHID;
    const float* wp = p_w2 + k * (3 * PHID) + q * PHID;
    float s = p_b2[k * 3 + q];
#pragma unroll 4
    for (int p = 0; p < PHID; ++p) s += hp[p] * wp[p];
    par[idx] = s;
}

// ---------------------------------------------------------------------------
// Per-b distribution params: mean = exp(p0), inv_scale = exp(-p1),
// weights = softmax over K of p2.  One thread per b.
// ---------------------------------------------------------------------------
__global__ __launch_bounds__(256)
void dist_params(const float* __restrict__ par,
                 float* __restrict__ weights_out,
                 float* __restrict__ dp) {
    const int b = blockIdx.x * blockDim.x + threadIdx.x;    // < B
    const float* p = par + b * 15;
    float wl[KMIX];
    float mx = -3.4e38f;
#pragma unroll
    for (int k = 0; k < KMIX; ++k) { wl[k] = p[k * 3 + 2]; mx = fmaxf(mx, wl[k]); }
    float sum = 0.0f;
#pragma unroll
    for (int k = 0; k < KMIX; ++k) { wl[k] = __expf(wl[k] - mx); sum += wl[k]; }
    const float inv = 1.0f / sum;
#pragma unroll
    for (int k = 0; k < KMIX; ++k) {
        const float wk = wl[k] * inv;
        weights_out[b * KMIX + k] = wk;
        dp[b * 15 + k * 3 + 0] = __expf(p[k * 3 + 0]);    // mean
        dp[b * 15 + k * 3 + 1] = __expf(-p[k * 3 + 1]);   // 1/scale (exact)
        dp[b * 15 + k * 3 + 2] = wk;
    }
}

// ---------------------------------------------------------------------------
// Survival stage: probs[b,s] = sum_k w_k * (sig(z1) - sig(z2)),
// z1 = (s+0.5-mean)*isc, z2 = z1 - isc, sig(z) = 1/(1+exp(z)).
// (the +1e-6 in f1 and f2 cancels exactly in f1-f2)
// Grid: B*8 blocks of 256 threads; thread -> 4 consecutive s (float4 store).
// ---------------------------------------------------------------------------
__global__ __launch_bounds__(256)
void survival_probs(const float* __restrict__ dp,
                    float* __restrict__ probs) {
    const int b     = blockIdx.x >> 3;
    const int chunk = blockIdx.x & 7;
    const int s0    = chunk * 1024 + threadIdx.x * 4;

    float mean[KMIX], isc[KMIX], wk[KMIX];
    const float* p = dp + b * 15;
#pragma unroll
    for (int k = 0; k < KMIX; ++k) {
        mean[k] = p[k * 3 + 0];
        isc[k]  = p[k * 3 + 1];
        wk[k]   = p[k * 3 + 2];
    }

    float4 out;
#pragma unroll
    for (int u = 0; u < 4; ++u) {
        const float s = (float)(s0 + u);
        float acc = 0.0f;
#pragma unroll
        for (int k = 0; k < KMIX; ++k) {
            const float z1 = (s + 0.5f - mean[k]) * isc[k];
            const float z2 = z1 - isc[k];
            const float f1 = 1.0f / (1.0f + __expf(z1));
            const float f2 = 1.0f / (1.0f + __expf(z2));
            acc += wk[k] * (f1 - f2);
        }
        (&out.x)[u] = acc;
    }
    *(float4*)(probs + b * SLEN + s0) = out;
}

// ---------------------------------------------------------------------------
// Host launcher
// ---------------------------------------------------------------------------
extern "C" void kernel_launch(void* const* d_in, const int* in_sizes, int n_in,
                              void* d_out, int out_size, void* d_ws, size_t ws_size,
                              hipStream_t stream) {
    const float* input = (const float*)d_in[0];  // (B,1,512)
    const float* w_ih  = (const float*)d_in[1];  // (1536,512)
    // d_in[2] = w_hh : dead in the reference (h0 == 0)
    const float* b_ih  = (const float*)d_in[3];  // (1536)
    const float* b_hh  = (const float*)d_in[4];  // (1536)
    const float* p_w1  = (const float*)d_in[5];  // (5,256,512) -> W[N=1280][512]
    const float* p_b1  = (const float*)d_in[6];  // (1280)
    const float* p_w2  = (const float*)d_in[7];  // (5,3,256)
    const float* p_b2  = (const float*)d_in[8];  // (5,3)
    // d_in[9] = enc_seq_len = 8192 (compile-time constant SLEN)

    float* out    = (float*)d_out;
    float* probs  = out;                                   // B*S
    float* state  = out + (size_t)BDIM * SLEN;             // B*512
    float* wout   = state + (size_t)BDIM * HDIM;           // B*5

    float* ws  = (float*)d_ws;
    float* gi  = ws;                                        // B*1536 floats
    float* hid = ws;                                        // B*1280 (aliases gi; gi is dead by then)
    float* par = ws + (size_t)BDIM * 3 * HDIM;              // B*15
    float* dp  = par + (size_t)BDIM * 15;                   // B*15

    // 1) gi = x @ w_ih^T + b_ih          (M=1024, N=1536, K=512)
    gemm_f32_wmma<<<dim3(1536 / 64, BDIM / 64), 256, 0, stream>>>(
        input, w_ih, b_ih, gi, 1536, 0);

    // 2) GRU gates -> h (== state output)
    gru_gate<<<(BDIM * HDIM) / 256, 256, 0, stream>>>(gi, b_hh, state);

    // 3) hid = tanh(h @ p_w1^T + p_b1)   (M=1024, N=1280, K=512)
    gemm_f32_wmma<<<dim3(1280 / 64, BDIM / 64), 256, 0, stream>>>(
        state, p_w1, p_b1, hid, 1280, 1);

    // 4) params[b,k,q]
    proj_params<<<(BDIM * 15) / 256, 256, 0, stream>>>(hid, p_w2, p_b2, par);

    // 5) means / inv_scales / softmax weights
    dist_params<<<BDIM / 256, 256, 0, stream>>>(par, wout, dp);

    // 6) survival probabilities
    survival_probs<<<BDIM * 8, 256, 0, stream>>>(dp, probs);
}